// RegionProposalNetwork_48713519072063
// MI455X (gfx1250) — compile-verified
//
#include <hip/hip_runtime.h>
#include <hip/hip_bf16.h>
#include <math.h>

// ---------------- problem constants (from reference) ----------------
#define RPN_B      16
#define RPN_TOTAL  261888
#define RPN_KTOT   4768      // sum of per-level min(1000, n)
#define RPN_POST   1000
#define RPN_CHUNK  4096      // floats per TDM chunk (16KB)
#define RPN_NBIN   4096

typedef unsigned int u32x4 __attribute__((ext_vector_type(4)));
typedef int          i32x8 __attribute__((ext_vector_type(8)));
typedef int          i32x4 __attribute__((ext_vector_type(4)));

#if __has_builtin(__builtin_amdgcn_tensor_load_to_lds) && __has_builtin(__builtin_amdgcn_s_wait_tensorcnt)
#define RPN_HAVE_TDM 1
#else
#define RPN_HAVE_TDM 0
#endif

// order-preserving float -> uint key (descending sort == descending uint)
__device__ __forceinline__ unsigned rpn_key(float f) {
  unsigned u = __float_as_uint(f);
  return (u & 0x80000000u) ? ~u : (u | 0x80000000u);
}

#if RPN_HAVE_TDM
// Issue one TDM descriptor: 1 row of `nelem` f32 from global -> LDS byte offset.
// D# layout per CDNA5 ISA (count=1, type=2,
// data_size=4B, tile_dim0=nelem, tile_dim1=1, no pad/iterate/multicast).
// This toolchain exposes the 6-arg builtin:
//   (uint32x4 g0, int32x8 g1, int32x4 g2, int32x4 g3, int32x8 extra, i32 cpol)
__device__ __forceinline__ void rpn_tdm_load(unsigned lds_byte, const void* gp, unsigned nelem) {
  unsigned long long ga = (unsigned long long)(size_t)gp;
  u32x4 g0;
  g0[0] = 1u;                                              // count=1 (valid), user mode
  g0[1] = lds_byte;                                        // lds_addr
  g0[2] = (unsigned)ga;                                    // global_addr[31:0]
  g0[3] = (unsigned)((ga >> 32) & 0x01FFFFFFull) | (2u << 30); // addr[56:32] | type=2
  i32x8 g1;
  g1[0] = (int)(2u << 16);                                 // data_size=4B, mask=0
  g1[1] = (int)((nelem & 0xFFFFu) << 16);                  // tensor_dim0[15:0]
  g1[2] = (int)(((nelem >> 16) & 0xFFFFu) | (1u << 16));   // tensor_dim0[31:16], tensor_dim1=1
  g1[3] = (int)((nelem & 0xFFFFu) << 16);                  // tile_dim0=nelem
  g1[4] = 1;                                               // tile_dim1=1, tile_dim2=0
  g1[5] = (int)nelem;                                      // tensor_dim0_stride (single row)
  g1[6] = 0;
  g1[7] = 0;
  i32x4 z4 = {0, 0, 0, 0};                                 // groups 2/3 unused (<=2D)
  i32x8 z8 = {0, 0, 0, 0, 0, 0, 0, 0};
  __builtin_amdgcn_tensor_load_to_lds(g0, g1, z4, z4, z8, 0);
}
#endif

// Stream `n` f32 from global through LDS in chunks, double-buffered via the
// Tensor Data Mover; wave 0 drives the DMA + s_wait_tensorcnt, all waves consume.
template <typename F>
__device__ void rpn_stream(const float* __restrict__ g, int n, float* buf, F&& consume) {
  const int tid = threadIdx.x;
  const int nth = blockDim.x;
  const int nch = (n + RPN_CHUNK - 1) / RPN_CHUNK;
#if RPN_HAVE_TDM
  if (tid < 32) {
    int r0 = n < RPN_CHUNK ? n : RPN_CHUNK;
    rpn_tdm_load(0u, g, (unsigned)r0);                     // prefetch chunk 0 -> buf0
  }
  for (int c = 0; c < nch; ++c) {
    int rem = n - c * RPN_CHUNK; if (rem > RPN_CHUNK) rem = RPN_CHUNK;
    if (tid < 32) {
      if (c + 1 < nch) {
        int rn = n - (c + 1) * RPN_CHUNK; if (rn > RPN_CHUNK) rn = RPN_CHUNK;
        rpn_tdm_load((unsigned)(((c + 1) & 1) * RPN_CHUNK * 4), g + (c + 1) * RPN_CHUNK, (unsigned)rn);
        __builtin_amdgcn_s_wait_tensorcnt(1);              // oldest (chunk c) complete
      } else {
        __builtin_amdgcn_s_wait_tensorcnt(0);              // drain
      }
    }
    __syncthreads();                                       // chunk c visible to all waves
    consume(buf + (c & 1) * RPN_CHUNK, rem, c * RPN_CHUNK);
    __syncthreads();                                       // done before buffer reuse
  }
#else
  for (int c = 0; c < nch; ++c) {
    int rem = n - c * RPN_CHUNK; if (rem > RPN_CHUNK) rem = RPN_CHUNK;
    for (int i = tid; i < rem; i += nth) buf[i] = g[c * RPN_CHUNK + i];
    __syncthreads();
    consume(buf, rem, c * RPN_CHUNK);
    __syncthreads();
  }
#endif
}

// ---------------- kernel A: per-(image, level) exact top-k via 2-level radix select ----
__global__ void rpn_topk_kernel(const float* __restrict__ objectness, int* __restrict__ topk) {
  constexpr int LOFF[5]  = {0, 196608, 245760, 258048, 261120};
  constexpr int LN[5]    = {196608, 49152, 12288, 3072, 768};
  constexpr int LK[5]    = {1000, 1000, 1000, 1000, 768};
  constexpr int LBASE[5] = {0, 1000, 2000, 3000, 4000};

  const int blk = blockIdx.x;
  const int b = blk / 5, l = blk % 5;
  const int n = LN[l], k = LK[l];
  const float* src = objectness + (size_t)b * RPN_TOTAL + LOFF[l];

  extern __shared__ unsigned char rpn_smemA[];
  float*    buf  = (float*)rpn_smemA;                                  // 2 * CHUNK f32
  unsigned* hist = (unsigned*)(rpn_smemA + 2 * RPN_CHUNK * sizeof(float));
  int*      ctrl = (int*)(rpn_smemA + 2 * RPN_CHUNK * sizeof(float) + RPN_NBIN * sizeof(unsigned));
  const int tid = threadIdx.x;
  const int nth = blockDim.x;

  // pass 1: coarse histogram over key[31:20]
  for (int i = tid; i < RPN_NBIN; i += nth) hist[i] = 0u;
  __syncthreads();
  rpn_stream(src, n, buf, [&](const float* cb, int rem, int base) {
    (void)base;
    for (int i = tid; i < rem; i += nth)
      atomicAdd(&hist[rpn_key(cb[i]) >> 20], 1u);
  });
  if (tid == 0) {
    int cum = 0, t = 0, chi = 0;
    for (int bin = RPN_NBIN - 1; bin >= 0; --bin) {
      int c = (int)hist[bin];
      if (cum + c >= k) { t = bin; chi = cum; break; }
      cum += c;
      if (bin == 0) { t = 0; chi = cum; }
    }
    ctrl[0] = t; ctrl[1] = chi;
  }
  __syncthreads();
  const unsigned t1 = (unsigned)ctrl[0];
  const int chi1 = ctrl[1];
  const int k2 = k - chi1;
  __syncthreads();

  // pass 2: refine inside threshold bin over key[19:8]
  for (int i = tid; i < RPN_NBIN; i += nth) hist[i] = 0u;
  __syncthreads();
  rpn_stream(src, n, buf, [&](const float* cb, int rem, int base) {
    (void)base;
    for (int i = tid; i < rem; i += nth) {
      unsigned u = rpn_key(cb[i]);
      if ((u >> 20) == t1) atomicAdd(&hist[(u >> 8) & 0xFFFu], 1u);
    }
  });
  if (tid == 0) {
    int cum = 0, t = 0, chi = 0;
    for (int bin = RPN_NBIN - 1; bin >= 0; --bin) {
      int c = (int)hist[bin];
      if (cum + c >= k2) { t = bin; chi = cum; break; }
      cum += c;
      if (bin == 0) { t = 0; chi = cum; }
    }
    ctrl[2] = t; ctrl[3] = chi;
    ctrl[4] = 0; ctrl[5] = 0; ctrl[6] = 0;                 // compaction counters
  }
  __syncthreads();
  const unsigned t2 = (unsigned)ctrl[2];
  const int chi2 = ctrl[3];
  const int r2 = k2 - chi2;

  // pass 3: compaction -> exactly k global indices
  int* outp = topk + (size_t)b * RPN_KTOT + LBASE[l];
  const int goff = LOFF[l];
  rpn_stream(src, n, buf, [&](const float* cb, int rem, int base) {
    for (int i = tid; i < rem; i += nth) {
      unsigned u = rpn_key(cb[i]);
      unsigned b1 = u >> 20;
      int slot = -1;
      if (b1 > t1) {
        slot = atomicAdd(&ctrl[4], 1);
      } else if (b1 == t1) {
        unsigned b2 = (u >> 8) & 0xFFFu;
        if (b2 > t2) {
          slot = chi1 + atomicAdd(&ctrl[5], 1);
        } else if (b2 == t2) {
          int p = atomicAdd(&ctrl[6], 1);
          if (p < r2) slot = chi1 + chi2 + p;
        }
      }
      if (slot >= 0) outp[slot] = goff + base + i;
    }
  });
}

// ---------------- kernel B: per-image gather + decode + clip + greedy NMS ----------------
__global__ void rpn_nms_kernel(const float* __restrict__ objectness,
                               const float* __restrict__ deltas,
                               const float* __restrict__ anchors,
                               const int* __restrict__ topk,
                               float* __restrict__ out) {
  const int b = blockIdx.x;
  const int tid = threadIdx.x;
  const int nth = blockDim.x;        // 1024 = 32 wave32 waves
  const int K = RPN_KTOT;

  extern __shared__ unsigned char rpn_smemB[];
  float* sc  = (float*)rpn_smemB;    // scores (-1 == suppressed/invalid)
  float* bx0 = sc  + K;
  float* by0 = bx0 + K;
  float* bx1 = by0 + K;
  float* by1 = bx1 + K;
  float* slv = by1 + K;              // FPN level as float
  float* rval = slv + K;             // 1024 reduction values
  int*   ridx = (int*)(rval + 1024); // 1024 reduction indices

  const float CLIPLOG = 4.135166556742356f;   // log(1000/16)

  // gather + sigmoid + decode + clip + validity
  for (int i = tid; i < K; i += nth) {
    int gi = topk[(size_t)b * K + i];
    float o = objectness[(size_t)b * RPN_TOTAL + gi];
    float score = 1.f / (1.f + __expf(-o));
    float ax0 = anchors[4 * gi + 0], ay0 = anchors[4 * gi + 1];
    float ax1 = anchors[4 * gi + 2], ay1 = anchors[4 * gi + 3];
    const float* dp = deltas + ((size_t)b * RPN_TOTAL + gi) * 4;
    float dx = dp[0], dy = dp[1];
    float dw = fminf(dp[2], CLIPLOG), dh = fminf(dp[3], CLIPLOG);
    float w = ax1 - ax0, h = ay1 - ay0;
    float cx = ax0 + 0.5f * w, cy = ay0 + 0.5f * h;
    float pcx = dx * w + cx, pcy = dy * h + cy;
    float pw = __expf(dw) * w, ph = __expf(dh) * h;
    float x0 = fminf(fmaxf(pcx - 0.5f * pw, 0.f), 1024.f);
    float y0 = fminf(fmaxf(pcy - 0.5f * ph, 0.f), 1024.f);
    float x1 = fminf(fmaxf(pcx + 0.5f * pw, 0.f), 1024.f);
    float y1 = fminf(fmaxf(pcy + 0.5f * ph, 0.f), 1024.f);
    bool valid = ((x1 - x0) >= 1e-3f) && ((y1 - y0) >= 1e-3f) && (score >= 0.f);
    bx0[i] = x0; by0[i] = y0; bx1[i] = x1; by1[i] = y1;
    sc[i]  = valid ? score : -1.f;
    slv[i] = (gi < 196608) ? 0.f : (gi < 245760) ? 1.f
           : (gi < 258048) ? 2.f : (gi < 261120) ? 3.f : 4.f;
  }
  __syncthreads();

  // level offset = max(clipped coords) + 1  (block max reduction)
  float m = 0.f;
  for (int i = tid; i < K; i += nth)
    m = fmaxf(m, fmaxf(fmaxf(bx0[i], by0[i]), fmaxf(bx1[i], by1[i])));
  rval[tid] = m;
  __syncthreads();
  for (int s = nth >> 1; s > 0; s >>= 1) {
    if (tid < s) rval[tid] = fmaxf(rval[tid], rval[tid + s]);
    __syncthreads();
  }
  const float lvloff = rval[0] + 1.f;
  __syncthreads();

  float* outb = out + (size_t)b * RPN_POST * 5;
  for (int it = 0; it < RPN_POST; ++it) {
    // block argmax, first-index tie-break (matches jnp.argmax)
    float bv = -2.f; int bi = K;
    for (int i = tid; i < K; i += nth) {
      float v = sc[i];
      if (v > bv) { bv = v; bi = i; }
    }
    rval[tid] = bv; ridx[tid] = bi;
    __syncthreads();
    for (int s = nth >> 1; s > 0; s >>= 1) {
      if (tid < s) {
        float v2 = rval[tid + s]; int i2 = ridx[tid + s];
        if (v2 > rval[tid] || (v2 == rval[tid] && i2 < ridx[tid])) { rval[tid] = v2; ridx[tid] = i2; }
      }
      __syncthreads();
    }
    const int j = ridx[0];
    const float sj = rval[0];
    __syncthreads();

    if (sj <= 0.f) {                 // exhausted: remaining rows are zeros
      int remain = (RPN_POST - it) * 5;
      float* z = outb + (size_t)it * 5;
      for (int e = tid; e < remain; e += nth) z[e] = 0.f;
      break;
    }

    // suppress IoU > 0.7 against the winner, on level-offset boxes
    float wo  = slv[j] * lvloff;
    float wx0 = bx0[j] + wo, wy0 = by0[j] + wo;
    float wx1 = bx1[j] + wo, wy1 = by1[j] + wo;
    float wa  = (wx1 - wx0) * (wy1 - wy0);
    for (int i = tid; i < K; i += nth) {
      float o2  = slv[i] * lvloff;
      float cx0 = bx0[i] + o2, cy0 = by0[i] + o2;
      float cx1 = bx1[i] + o2, cy1 = by1[i] + o2;
      float ltx = fmaxf(wx0, cx0), lty = fmaxf(wy0, cy0);
      float rbx = fminf(wx1, cx1), rby = fminf(wy1, cy1);
      float iw = fmaxf(rbx - ltx, 0.f), ih = fmaxf(rby - lty, 0.f);
      float inter = iw * ih;
      float a2 = (cx1 - cx0) * (cy1 - cy0);
      float iou = inter / (wa + a2 - inter);
      if (iou > 0.7f) sc[i] = -1.f;
    }
    if (tid == 0) {
      sc[j] = -1.f;
      float* o5 = outb + (size_t)it * 5;
      o5[0] = bx0[j]; o5[1] = by0[j]; o5[2] = bx1[j]; o5[3] = by1[j]; o5[4] = sj;
    }
    __syncthreads();
  }
}

// ---------------- launch ----------------
extern "C" void kernel_launch(void* const* d_in, const int* in_sizes, int n_in,
                              void* d_out, int out_size, void* d_ws, size_t ws_size,
                              hipStream_t stream) {
  (void)in_sizes; (void)n_in; (void)out_size; (void)ws_size;
  const float* objectness = (const float*)d_in[0];
  const float* deltas     = (const float*)d_in[1];
  const float* anchors    = (const float*)d_in[2];
  float* out = (float*)d_out;
  int* topk = (int*)d_ws;                         // [B][4768] selected indices

  size_t shmA = 2 * RPN_CHUNK * sizeof(float)     // double-buffered TDM staging
              + RPN_NBIN * sizeof(unsigned)       // histogram
              + 64;                               // control
  rpn_topk_kernel<<<RPN_B * 5, 256, shmA, stream>>>(objectness, topk);

  size_t shmB = (size_t)(6 * RPN_KTOT + 1024) * sizeof(float) + 1024 * sizeof(int);
  rpn_nms_kernel<<<RPN_B, 1024, shmB, stream>>>(objectness, deltas, anchors, topk, out);
}